// GPlightPredictor_704374636700
// MI455X (gfx1250) — compile-verified
//
#include <hip/hip_runtime.h>
#include <hip/hip_bf16.h>

typedef __attribute__((ext_vector_type(2))) float v2f;
typedef __attribute__((ext_vector_type(8))) float v8f;

// ---------------------------------------------------------------------------
// Degree / normalization kernels
// ---------------------------------------------------------------------------
__global__ void k_deg_init(float* __restrict__ deg, int N) {
    int i = blockIdx.x * blockDim.x + threadIdx.x;
    if (i < N) deg[i] = 1.0f;  // self-loop contributes 1 to in-degree
}

__global__ void k_deg_accum(const int* __restrict__ dst, float* __restrict__ deg, int E) {
    int e = blockIdx.x * blockDim.x + threadIdx.x;
    if (e < E) atomicAdd(&deg[dst[e]], 1.0f);
}

__global__ void k_deg_rsqrt(float* __restrict__ deg, int N) {
    int i = blockIdx.x * blockDim.x + threadIdx.x;
    if (i < N) deg[i] = rsqrtf(deg[i]);  // deg >= 1 always (self-loops)
}

// ---------------------------------------------------------------------------
// Pad B[K,NB] (row-major) into Bp[K,16] with zero columns NB..15.
// Removes all per-lane conditionals from the WMMA B-fragment loads.
// ---------------------------------------------------------------------------
__global__ void k_pad_b(const float* __restrict__ B, float* __restrict__ Bp,
                        int K, int NB) {
    int t = blockIdx.x * blockDim.x + threadIdx.x;
    if (t < K * 16) {
        int k = t >> 4, c = t & 15;
        Bp[t] = (c < NB) ? B[k * NB + c] : 0.0f;
    }
}

// ---------------------------------------------------------------------------
// WMMA fp32 GEMM: D[N, NBOUT] = (relu?)A[N, K] * Bp[K, 16]
// One wave per 16-row tile, V_WMMA_F32_16X16X4_F32 over the K dimension.
//   A fragment (16x4 f32): lanes 0-15 hold K=k0,k0+1 ; lanes 16-31 hold k0+2,k0+3
//   B fragment (4x16 f32): lanes 0-15 cols (K=k0,k0+1); lanes 16-31 cols (k0+2,k0+3)
//   C/D (16x16 f32): VGPR r -> row r (lanes 0-15) / r+8 (lanes 16-31), col = lane&15
// No divergence inside the K-loop: B is pre-padded to 16 cols, RELU is a
// template parameter, ragged-row loads are clamped (EXEC stays all-ones).
// ---------------------------------------------------------------------------
template <bool RELU>
__global__ void k_gemm_wmma_f32(const float* __restrict__ A,
                                const float* __restrict__ Bp,  // [K,16] padded
                                float* __restrict__ D,
                                int nrows, int K, int NBOUT) {
    const int lane = threadIdx.x & 31;
    const int wave = threadIdx.x >> 5;
    const int tile = blockIdx.x * (blockDim.x >> 5) + wave;
    const int row0 = tile * 16;
    if (row0 >= nrows) return;  // wave-uniform exit: EXEC all-ones for WMMA

    const bool hi = lane >= 16;
    const int  lr = lane & 15;
    const int arow = min(row0 + lr, nrows - 1);  // clamp ragged tail, keep EXEC full

    const float* ap = A + (size_t)arow * (size_t)K + (hi ? 2 : 0);
    const float* bp = Bp + (hi ? 32 : 0) + lr;   // element B[kb*16 + lr], kb=k0+(hi?2:0)

    v8f acc = {};
    #pragma unroll 4
    for (int k0 = 0; k0 < K; k0 += 4) {
        float a0 = ap[0];
        float a1 = ap[1];
        if (RELU) { a0 = fmaxf(a0, 0.0f); a1 = fmaxf(a1, 0.0f); }
        v2f a; a[0] = a0; a[1] = a1;
        v2f b; b[0] = bp[0]; b[1] = bp[16];
        // D = A x B + C  (neg_a, A, neg_b, B, c_mod, C, reuse_a, reuse_b)
        acc = __builtin_amdgcn_wmma_f32_16x16x4_f32(false, a, false, b,
                                                    (short)0, acc, false, false);
        ap += 4;
        bp += 64;
    }

    if (lr < NBOUT) {
        const int rbase = row0 + (hi ? 8 : 0);
        #pragma unroll
        for (int r = 0; r < 8; ++r) {
            const int row = rbase + r;
            if (row < nrows) D[(size_t)row * NBOUT + lr] = acc[r];
        }
    }
}

// ---------------------------------------------------------------------------
// Aggregation init: out[i,f] = H[i,f] * dis[i]^2 + bias[f]
// (self-loop term: norm = dis[i]^2; bias added pre-relu as in reference)
// ---------------------------------------------------------------------------
__global__ void k_agg_init(const float* __restrict__ H, const float* __restrict__ dis,
                           const float* __restrict__ bias, float* __restrict__ out,
                           int N, int fshift) {
    int t = blockIdx.x * blockDim.x + threadIdx.x;
    if (t < (N << fshift)) {
        int i = t >> fshift, f = t & ((1 << fshift) - 1);
        float s = dis[i];
        out[t] = H[t] * s * s + bias[f];
    }
}

// ---------------------------------------------------------------------------
// Edge scatter-add: out[dst] += H[src] * (dis[src]*dis[dst])
// One thread per (edge, float4 chunk): coalesced 128b gather, 4 f32 atomics.
// H/out tiles (<= 6.4 MB) live in the 192 MB L2, so atomics resolve in L2.
// cshift = log2(F/4) -> pure shift/mask indexing, no integer division.
// ---------------------------------------------------------------------------
__global__ void k_agg_edges(const int* __restrict__ src, const int* __restrict__ dst,
                            const float* __restrict__ dis, const float* __restrict__ H,
                            float* __restrict__ out, long long nthreads, int cshift) {
    long long t = (long long)blockIdx.x * blockDim.x + threadIdx.x;
    if (t >= nthreads) return;
    int e = (int)(t >> cshift);
    int c = ((int)t & ((1 << cshift) - 1)) << 2;
    int F = 4 << cshift;
    int s = src[e], d = dst[e];
    float w = dis[s] * dis[d];
    const float4 hv = *(const float4*)(H + (size_t)s * F + c);
    float* o = out + (size_t)d * F + c;
    atomicAdd(o + 0, hv.x * w);
    atomicAdd(o + 1, hv.y * w);
    atomicAdd(o + 2, hv.z * w);
    atomicAdd(o + 3, hv.w * w);
}

// ---------------------------------------------------------------------------
// Launcher
// ---------------------------------------------------------------------------
extern "C" void kernel_launch(void* const* d_in, const int* in_sizes, int n_in,
                              void* d_out, int out_size, void* d_ws, size_t ws_size,
                              hipStream_t stream) {
    const float* x  = (const float*)d_in[0];
    const int*   ei = (const int*)d_in[1];
    const float* W1 = (const float*)d_in[2];
    const float* b1 = (const float*)d_in[3];
    const float* W2 = (const float*)d_in[4];
    const float* b2 = (const float*)d_in[5];

    const int HID = in_sizes[3];               // 16
    const int C   = in_sizes[5];               // 8
    const int FIN = in_sizes[2] / HID;         // 128
    const int N   = in_sizes[0] / FIN;         // 100000
    const int E   = in_sizes[1] / 2;           // 3200000

    const int* src = ei;
    const int* dst = ei + E;

    float* ws   = (float*)d_ws;
    float* deg  = ws;                          // N   (becomes dis after rsqrt)
    float* H1   = deg  + (size_t)N;            // N*HID
    float* AGG1 = H1   + (size_t)N * HID;      // N*HID
    float* H2   = AGG1 + (size_t)N * HID;      // N*C
    float* B1p  = H2   + (size_t)N * C;        // FIN*16 (W1 padded to 16 cols)
    float* B2p  = B1p  + (size_t)FIN * 16;     // HID*16 (W2 padded to 16 cols)
    float* out  = (float*)d_out;               // N*C

    const int T = 256;
    const int wavesPerBlock = T / 32;
    const int tiles  = (N + 15) / 16;
    const int gTiles = (tiles + wavesPerBlock - 1) / wavesPerBlock;

    // log2 of per-edge float4 chunks (HID=16 -> 2, C=8 -> 1)
    int s1 = 0; while ((4 << (s1 + 1)) <= HID) ++s1;
    int s2 = 0; while ((4 << (s2 + 1)) <= C)   ++s2;

    // --- degree / normalization ---
    k_deg_init <<<(N + T - 1) / T, T, 0, stream>>>(deg, N);
    k_deg_accum<<<(E + T - 1) / T, T, 0, stream>>>(dst, deg, E);
    k_deg_rsqrt<<<(N + T - 1) / T, T, 0, stream>>>(deg, N);

    // --- pad weight matrices to 16 columns for branch-free B fragments ---
    k_pad_b<<<(FIN * 16 + T - 1) / T, T, 0, stream>>>(W1, B1p, FIN, HID);
    k_pad_b<<<(HID * 16 + T - 1) / T, T, 0, stream>>>(W2, B2p, HID, C);

    // --- layer 1: H1 = X @ W1 (WMMA f32) ---
    k_gemm_wmma_f32<false><<<gTiles, T, 0, stream>>>(x, B1p, H1, N, FIN, HID);

    // --- layer 1 aggregation: AGG1 = D^-1/2 (A+I) D^-1/2 H1 + b1 ---
    k_agg_init<<<((long long)N * HID + T - 1) / T, T, 0, stream>>>(H1, deg, b1, AGG1, N, s1 + 2);
    {
        long long nt = (long long)E << s1;
        k_agg_edges<<<(unsigned)((nt + T - 1) / T), T, 0, stream>>>(src, dst, deg, H1, AGG1, nt, s1);
    }

    // --- layer 2: H2 = relu(AGG1) @ W2 (WMMA f32, relu fused into A load) ---
    k_gemm_wmma_f32<true><<<gTiles, T, 0, stream>>>(AGG1, B2p, H2, N, HID, C);

    // --- layer 2 aggregation into d_out ---
    k_agg_init<<<((long long)N * C + T - 1) / T, T, 0, stream>>>(H2, deg, b2, out, N, s2 + 2);
    {
        long long nt = (long long)E << s2;
        k_agg_edges<<<(unsigned)((nt + T - 1) / T), T, 0, stream>>>(src, dst, deg, H2, out, nt, s2);
    }
}